// MPNN_82420422410589
// MI455X (gfx1250) — compile-verified
//
#include <hip/hip_runtime.h>
#include <hip/hip_bf16.h>

// ---------------------------------------------------------------------------
// MPNN forward on MI455X (gfx1250).
// All dense layers use V_WMMA_F32_16X16X4_F32 (fp32, matches reference dtype).
// Edge aggregation = gather + global_atomic_add_f32 (L2-resident: working set
// ~64MB << 192MB L2). GEMM K-loops fully unrolled via template params.
// ---------------------------------------------------------------------------

typedef float v2f __attribute__((ext_vector_type(2)));
typedef float v8f __attribute__((ext_vector_type(8)));

#define FEAT 64
#define NGRAPHS 50

__device__ __forceinline__ v8f wmma4(v2f a, v2f b, v8f c) {
  // 8 args: (neg_a, A, neg_b, B, c_mod, C, reuse_a, reuse_b)
  return __builtin_amdgcn_wmma_f32_16x16x4_f32(false, a, false, b, (short)0, c,
                                               false, false);
}

// ---------------------------------------------------------------------------
// deg[col[e]] += 1
__global__ void deg_kernel(const int* __restrict__ col, float* __restrict__ deg,
                           int E) {
  int e = blockIdx.x * blockDim.x + threadIdx.x;
  if (e < E) atomicAdd(&deg[col[e]], 1.0f);
}

// deg_inv = 1/deg ; degmax via block reduce + atomicMax on float bits (>0)
__global__ void deginv_max_kernel(const float* __restrict__ deg,
                                  float* __restrict__ deg_inv,
                                  unsigned* __restrict__ degmax, int n) {
  __shared__ float smax[256];
  int i = blockIdx.x * blockDim.x + threadIdx.x;
  float d = 0.0f;
  if (i < n) {
    d = deg[i];
    deg_inv[i] = 1.0f / d;
  }
  smax[threadIdx.x] = d;
  __syncthreads();
  for (int s = 128; s > 0; s >>= 1) {
    if (threadIdx.x < s)
      smax[threadIdx.x] = fmaxf(smax[threadIdx.x], smax[threadIdx.x + s]);
    __syncthreads();
  }
  if (threadIdx.x == 0) atomicMax(degmax, __float_as_uint(smax[0]));
}

// ---------------------------------------------------------------------------
// cur = relu(x @ W_init [4,64]), h = relu(x @ W_e1 [4,63] zero-padded to 64)
// One wave per 16-node tile; K=4 -> single WMMA per 16x16 output tile.
__global__ void init_embed_kernel(const float* __restrict__ x,
                                  const float* __restrict__ Wi,
                                  const float* __restrict__ We1,
                                  float* __restrict__ cur,
                                  float* __restrict__ h, int ntiles) {
  int lane = threadIdx.x & 31;
  int wave = threadIdx.x >> 5;
  int tile = blockIdx.x * (blockDim.x >> 5) + wave;
  if (tile >= ntiles) return;
  int mlo = lane & 15, hi = lane >> 4;
  int m = tile * 16 + mlo;
  int kbo = hi * 2;  // A 16x4 f32: lanes<16 hold K{0,1}, lanes>=16 hold K{2,3}

  v2f a = *(const v2f*)(x + m * 4 + kbo);

#pragma unroll
  for (int nt = 0; nt < 4; ++nt) {
    int n = nt * 16 + mlo;
    v2f b;
    b.x = Wi[kbo * 64 + n];
    b.y = Wi[(kbo + 1) * 64 + n];
    v8f c = {};
    c = wmma4(a, b, c);
#pragma unroll
    for (int v = 0; v < 8; ++v)
      cur[(size_t)(tile * 16 + v + hi * 8) * FEAT + n] = fmaxf(c[v], 0.0f);
  }
#pragma unroll
  for (int nt = 0; nt < 4; ++nt) {
    int n = nt * 16 + mlo;
    v2f b;
    b.x = (n < 63) ? We1[kbo * 63 + n] : 0.0f;
    b.y = (n < 63) ? We1[(kbo + 1) * 63 + n] : 0.0f;
    v8f c = {};
    c = wmma4(a, b, c);
#pragma unroll
    for (int v = 0; v < 8; ++v)
      h[(size_t)(tile * 16 + v + hi * 8) * FEAT + n] = fmaxf(c[v], 0.0f);
  }
}

// ---------------------------------------------------------------------------
// dst[col[e], f] += src[row[e], f]   (thread = one (edge, feature))
__global__ void scatter_agg_kernel(const float* __restrict__ src,
                                   const int* __restrict__ row,
                                   const int* __restrict__ col,
                                   float* __restrict__ dst, int E) {
  unsigned gid = blockIdx.x * blockDim.x + threadIdx.x;
  unsigned e = gid >> 6;
  if (e >= (unsigned)E) return;
  int f = gid & 63;
  atomicAdd(&dst[(size_t)col[e] * FEAT + f], src[(size_t)row[e] * FEAT + f]);
}

// ---------------------------------------------------------------------------
// out = relu( scale1(A1) @ W[0:K1] + A2 @ W[K1:K1+K2] )
//   YMODE: A1 col<63 -> A1[m,c]*deg_inv[m]; col 63 -> deg[m]/deg_max
//   SCALE1: apply per-row scale (deg_inv) to A1
// One wave per 16-row tile, full 16x64 output; K fully unrolled (step 4, WMMA).
template <int K1, int K2, bool YMODE, bool SCALE1>
__global__ void gemm_relu_kernel(const float* __restrict__ A1,
                                 const float* __restrict__ rowScale1,
                                 const float* __restrict__ A2,
                                 const float* __restrict__ W,
                                 float* __restrict__ out, int ntiles,
                                 const float* __restrict__ deg,
                                 const unsigned* __restrict__ degmaxBits) {
  int lane = threadIdx.x & 31;
  int wave = threadIdx.x >> 5;
  int tile = blockIdx.x * (blockDim.x >> 5) + wave;
  if (tile >= ntiles) return;
  int mlo = lane & 15, hi = lane >> 4;
  int m = tile * 16 + mlo;
  int kbo = hi * 2;

  float s1 = SCALE1 ? rowScale1[m] : 1.0f;
  float degTerm = 0.0f;
  if (YMODE) degTerm = deg[m] / __uint_as_float(*degmaxBits);

  v8f acc[4];
  acc[0] = v8f{};
  acc[1] = v8f{};
  acc[2] = v8f{};
  acc[3] = v8f{};

  const float* Arow = A1 + (size_t)m * FEAT;
#pragma unroll
  for (int k0 = 0; k0 < K1; k0 += 4) {
    int c0 = k0 + kbo;
    v2f a = *(const v2f*)(Arow + c0);  // 8B aligned (c0 even)
    if (SCALE1) {
      a.x *= s1;
      a.y *= s1;
    }
    if (YMODE && k0 == K1 - 4) {
      // only the last k-step can touch the deg/deg_max column (63)
      if (c0 == 63) a.x = degTerm;
      if (c0 + 1 == 63) a.y = degTerm;
    }
    const float* w0 = W + (size_t)c0 * 64 + mlo;
#pragma unroll
    for (int nt = 0; nt < 4; ++nt) {
      v2f b;
      b.x = w0[nt * 16];
      b.y = w0[64 + nt * 16];
      acc[nt] = wmma4(a, b, acc[nt]);
    }
  }
  if (K2 > 0) {
    const float* A2row = A2 + (size_t)m * FEAT;
    const float* W2 = W + (size_t)K1 * 64;
#pragma unroll
    for (int k0 = 0; k0 < K2; k0 += 4) {
      int c0 = k0 + kbo;
      v2f a = *(const v2f*)(A2row + c0);
      const float* w0 = W2 + (size_t)c0 * 64 + mlo;
#pragma unroll
      for (int nt = 0; nt < 4; ++nt) {
        v2f b;
        b.x = w0[nt * 16];
        b.y = w0[64 + nt * 16];
        acc[nt] = wmma4(a, b, acc[nt]);
      }
    }
  }

#pragma unroll
  for (int nt = 0; nt < 4; ++nt) {
#pragma unroll
    for (int v = 0; v < 8; ++v)
      out[(size_t)(tile * 16 + v + hi * 8) * FEAT + nt * 16 + mlo] =
          fmaxf(acc[nt][v], 0.0f);
  }
}

// ---------------------------------------------------------------------------
// per-graph mean of cur (block per graph; graphs are contiguous, equal size)
__global__ void pool_mean_kernel(const float* __restrict__ cur,
                                 float* __restrict__ pool, int npg) {
  __shared__ float sbuf[256];
  int g = blockIdx.x;
  int f = threadIdx.x & 63;
  int r = threadIdx.x >> 6;  // 0..3
  float s = 0.0f;
  for (int nd = r; nd < npg; nd += 4)
    s += cur[(size_t)(g * npg + nd) * FEAT + f];
  sbuf[threadIdx.x] = s;
  __syncthreads();
  if (r == 0)
    pool[g * FEAT + f] =
        (sbuf[f] + sbuf[64 + f] + sbuf[128 + f] + sbuf[192 + f]) / (float)npg;
}

// pooledW = relu(means @ W_pool)   (tiny [50,64]@[64,64])
__global__ void poolw_kernel(const float* __restrict__ pool,
                             const float* __restrict__ Wp,
                             float* __restrict__ pooledW) {
  int g = blockIdx.x, n = threadIdx.x;
  float s = 0.0f;
#pragma unroll 8
  for (int k = 0; k < FEAT; ++k) s += pool[g * FEAT + k] * Wp[k * FEAT + n];
  pooledW[g * FEAT + n] = fmaxf(s, 0.0f);
}

// out[n] = relu(pooledW[batch[n]]) . W_read[0:64] + relu(cur[n]) . W_read[64:128] + b
__global__ void readout_kernel(const float* __restrict__ pooledW,
                               const float* __restrict__ cur,
                               const int* __restrict__ batch,
                               const float* __restrict__ Wr,
                               const float* __restrict__ br,
                               float* __restrict__ out, int n) {
  int i = blockIdx.x * blockDim.x + threadIdx.x;
  if (i >= n) return;
  int g = batch[i];
  const float* pw = pooledW + (size_t)g * FEAT;  // already relu'd
  const float* cv = cur + (size_t)i * FEAT;      // already relu'd
  float s = br[0];
#pragma unroll 8
  for (int f = 0; f < FEAT; ++f) s += pw[f] * Wr[f];
#pragma unroll 8
  for (int f = 0; f < FEAT; ++f) s += cv[f] * Wr[64 + f];
  out[i] = s;
}

// ---------------------------------------------------------------------------
extern "C" void kernel_launch(void* const* d_in, const int* in_sizes, int n_in,
                              void* d_out, int out_size, void* d_ws,
                              size_t ws_size, hipStream_t stream) {
  const float* x      = (const float*)d_in[0];
  const float* W_init = (const float*)d_in[1];  // [4,64]
  const float* W_e1   = (const float*)d_in[2];  // [4,63]
  const float* W_e2   = (const float*)d_in[3];  // [64,64]
  const float* W_msg  = (const float*)d_in[4];  // [3,128,64]
  const float* W_upd  = (const float*)d_in[5];  // [3,128,64]
  const float* W_pool = (const float*)d_in[6];  // [64,64]
  const float* W_read = (const float*)d_in[7];  // [128,1]
  const float* b_read = (const float*)d_in[8];  // [1]
  const int* eidx     = (const int*)d_in[9];    // [2,E]
  const int* batch    = (const int*)d_in[10];   // [N]

  const int N = in_sizes[10];
  const int E = in_sizes[9] / 2;
  const int ntiles = N / 16;  // N = 50000 -> 3125 exact
  const int npg = N / NGRAPHS;

  const int* row = eidx;
  const int* col = eidx + E;

  // workspace carve-out (floats)
  float* ws = (float*)d_ws;
  size_t o = 0;
  float* deg     = ws + o; o += N;
  float* deg_inv = ws + o; o += N;
  float* hbuf    = ws + o; o += (size_t)N * FEAT;  // h, later msg
  float* aggY    = ws + o; o += (size_t)N * FEAT;  // agg / nagg (raw sums)
  float* cur0    = ws + o; o += (size_t)N * FEAT;
  float* cur1    = ws + o; o += (size_t)N * FEAT;
  float* eemb    = ws + o; o += (size_t)N * FEAT;
  float* pool    = ws + o; o += NGRAPHS * FEAT;
  float* pooledW = ws + o; o += NGRAPHS * FEAT;
  unsigned* degmax = (unsigned*)(ws + o); o += 1;

  const unsigned aggBlocks = (unsigned)(((size_t)E * 64 + 255) / 256);
  const int gemmBlocks = (ntiles + 7) / 8;  // 8 waves/block, 1 tile/wave

  // degrees
  hipMemsetAsync(deg, 0, (size_t)N * sizeof(float), stream);
  hipMemsetAsync(degmax, 0, sizeof(unsigned), stream);
  deg_kernel<<<(E + 255) / 256, 256, 0, stream>>>(col, deg, E);
  deginv_max_kernel<<<(N + 255) / 256, 256, 0, stream>>>(deg, deg_inv, degmax, N);

  // node init + edge embedding input
  init_embed_kernel<<<gemmBlocks, 256, 0, stream>>>(x, W_init, W_e1, cur0, hbuf,
                                                    ntiles);
  hipMemsetAsync(aggY, 0, (size_t)N * FEAT * sizeof(float), stream);
  scatter_agg_kernel<<<aggBlocks, 256, 0, stream>>>(hbuf, row, col, aggY, E);
  // edge_emb = relu(y @ W_e2), y built on the fly (deg_inv scale + deg/degmax col)
  gemm_relu_kernel<64, 0, true, true><<<gemmBlocks, 256, 0, stream>>>(
      aggY, deg_inv, nullptr, W_e2, eemb, ntiles, deg, degmax);

  // message-passing layers
  float* curIn = cur0;
  float* curOut = cur1;
  for (int l = 0; l < 3; ++l) {
    hipMemsetAsync(aggY, 0, (size_t)N * FEAT * sizeof(float), stream);
    scatter_agg_kernel<<<aggBlocks, 256, 0, stream>>>(curIn, row, col, aggY, E);
    // msg = relu([deg_inv*nagg | eemb] @ W_msg[l])
    gemm_relu_kernel<64, 64, false, true><<<gemmBlocks, 256, 0, stream>>>(
        aggY, deg_inv, eemb, W_msg + (size_t)l * 128 * 64, hbuf, ntiles,
        nullptr, nullptr);
    // cur = relu([cur | msg] @ W_upd[l])
    gemm_relu_kernel<64, 64, false, false><<<gemmBlocks, 256, 0, stream>>>(
        curIn, nullptr, hbuf, W_upd + (size_t)l * 128 * 64, curOut, ntiles,
        nullptr, nullptr);
    float* t = curIn; curIn = curOut; curOut = t;
  }

  // readout: f_pooled = (means @ W_pool)[batch]
  pool_mean_kernel<<<NGRAPHS, 256, 0, stream>>>(curIn, pool, npg);
  poolw_kernel<<<NGRAPHS, 64, 0, stream>>>(pool, W_pool, pooledW);
  readout_kernel<<<(N + 255) / 256, 256, 0, stream>>>(
      pooledW, curIn, batch, W_read, b_read, (float*)d_out, N);
}